// TensorRT_Vanilla_31731218382925
// MI455X (gfx1250) — compile-verified
//
#include <hip/hip_runtime.h>
#include <math.h>

// ---------------------------------------------------------------------------
// Problem constants (from the reference)
// ---------------------------------------------------------------------------
#define M_   16
#define N_   4096
#define D_   128
#define H_   32
#define P_   2048
#define CL_  4096
#define QKVC (3 * H_ * D_)        // 12288 columns of the fused QKV projection
#define KSPLIT 4                  // split-K factor for the QKV GEMM
#define ASPLIT 16                 // position splits per head in attention
#define CHUNKS (CL_ / 16)         // 256 sixteen-position chunks of the cache
#define PFDIST 64                 // prefetch distance in k-units (16 iters)

typedef __attribute__((ext_vector_type(2))) float v2f;
typedef __attribute__((ext_vector_type(8))) float v8f;

// ---------------------------------------------------------------------------
// Kernel 1: fused QKV projection, fp32 WMMA 16x16x4, split-K.
// One wave owns one 16x16 output tile for one K-slice of 1024.
//   A (16x4 f32): lane L (m = L&15, hh = L>>4): v0 = X[m][k+2hh], v1 = next
//   B (4x16 f32): lane L (n = L&15):            v0 = W[k+2hh][col0+n], v1 = row below
// W is streamed exactly once (192 MiB -> the dominant HBM cost); prefetch
// runs PFDIST ahead of the register pipeline to raise bytes-in-flight.
// ---------------------------------------------------------------------------
__global__ __launch_bounds__(256) void qkv_gemm_splitk(
    const float* __restrict__ X, const float* __restrict__ W,
    float* __restrict__ part) {
  const int lane = threadIdx.x & 31;
  const int wv   = threadIdx.x >> 5;
  const int gw   = blockIdx.x * 8 + wv;          // global wave id
  const int tile = gw % (QKVC / 16);             // 768 column tiles
  const int ks   = gw / (QKVC / 16);             // 4 K-splits
  const int col0 = tile * 16;
  const int n    = lane & 15;
  const int hh   = lane >> 4;

  const float* xrow = X + (size_t)n * N_;        // A: m == n (lane&15)
  const int k0 = ks * (N_ / KSPLIT);

  v8f c = {};
#pragma unroll 8
  for (int k = k0; k < k0 + N_ / KSPLIT; k += 4) {
    // speculative prefetch of the B rows this wave will load PFDIST ahead
    __builtin_prefetch(&W[(size_t)(k + PFDIST + 2 * hh) * QKVC + col0 + n], 0, 1);
    __builtin_prefetch(&W[(size_t)(k + PFDIST + 2 * hh + 1) * QKVC + col0 + n], 0, 1);

    v2f a = *(const v2f*)(xrow + k + 2 * hh);
    v2f b;
    b.x = W[(size_t)(k + 2 * hh) * QKVC + col0 + n];
    b.y = W[(size_t)(k + 2 * hh + 1) * QKVC + col0 + n];
    c = __builtin_amdgcn_wmma_f32_16x16x4_f32(false, a, false, b,
                                              (short)0, c, false, false);
  }

  // C/D layout: VGPR r -> row (r + 8*hh), lane&15 -> column
  float* out = part + (size_t)ks * M_ * QKVC + col0 + n;
#pragma unroll
  for (int r = 0; r < 8; ++r)
    out[(size_t)(r + 8 * hh) * QKVC] = c[r];
}

// ---------------------------------------------------------------------------
// Kernel 1b: reduce the split-K partials into the final qkv buffer.
// ---------------------------------------------------------------------------
__global__ __launch_bounds__(256) void qkv_reduce(
    const float* __restrict__ part, float* __restrict__ qkv) {
  const int i = blockIdx.x * 256 + threadIdx.x;  // 16*12288 = 196608 total
  float s = 0.f;
#pragma unroll
  for (int ks = 0; ks < KSPLIT; ++ks)
    s += part[(size_t)ks * M_ * QKVC + i];
  qkv[i] = s;
}

// ---------------------------------------------------------------------------
// Kernel 2: attention partials. One block = (head, position-split of 256).
// Flash-style: out_acc[m][d] += exp(S[m][p]) * V[p][d], denom[m] += exp(S).
// The chunk covering positions [P, P+16) reads K/V from the qkv buffer
// (the reference's cache overwrite) -- we never mutate the input caches.
// ---------------------------------------------------------------------------
__global__ __launch_bounds__(256) void attn_partial(
    const float* __restrict__ qkv, const float* __restrict__ cacheK,
    const float* __restrict__ cacheV, float* __restrict__ pacc,
    float* __restrict__ pden) {
  __shared__ float e_st[8 * 256];     // per-wave 16x16 exp(S) staging tile
  __shared__ float acc[M_ * D_];      // block-level numerator accumulator
  __shared__ float den[M_];           // block-level denominator accumulator

  const int h    = blockIdx.x / ASPLIT;
  const int sp   = blockIdx.x % ASPLIT;
  const int lane = threadIdx.x & 31;
  const int wv   = threadIdx.x >> 5;
  const int n    = lane & 15;
  const int hh   = lane >> 4;

  for (int i = threadIdx.x; i < M_ * D_; i += 256) acc[i] = 0.f;
  if (threadIdx.x < M_) den[threadIdx.x] = 0.f;
  __syncthreads();

  const float* qbase = qkv + h * D_;  // Q rows, stride QKVC
  float* ew = e_st + wv * 256;

  v8f af[8] = {};                     // 8 d-tiles of 16x16 fp32 accumulators
  float dacc = 0.f;

  const int cps = CHUNKS / ASPLIT;    // 16 chunks per split, 2 per wave
  for (int j = 0; j < cps / 8; ++j) {
    const int c    = sp * cps + wv * (cps / 8) + j;
    const int pos0 = c * 16;

    const float* kbase;
    const float* vbase;
    int stride;
    if (pos0 == P_) {                 // the freshly-written cache slice
      kbase  = qkv + 1 * H_ * D_ + h * D_;
      vbase  = qkv + 2 * H_ * D_ + h * D_;
      stride = QKVC;
    } else {
      kbase  = cacheK + ((size_t)h * CL_ + pos0) * D_;
      vbase  = cacheV + ((size_t)h * CL_ + pos0) * D_;
      stride = D_;
    }

    // ---- scores: S[16m x 16p] = sum_d q[m][d] * k[p][d]  (32 WMMAs) ----
    v8f s8 = {};
#pragma unroll 16
    for (int g = 0; g < 32; ++g) {
      v2f a = *(const v2f*)(qbase + (size_t)n * QKVC + 4 * g + 2 * hh);
      v2f b = *(const v2f*)(kbase + (size_t)n * stride + 4 * g + 2 * hh);
      s8 = __builtin_amdgcn_wmma_f32_16x16x4_f32(false, a, false, b,
                                                 (short)0, s8, false, false);
    }

    // ---- raw exp (reference does no max-subtraction), stage via LDS ----
#pragma unroll
    for (int r = 0; r < 8; ++r)
      ew[(r + 8 * hh) * 16 + n] = expf(s8[r]);

    // denominator partial: lane n (hh==0 lanes are authoritative)
    float ds = 0.f;
#pragma unroll
    for (int p = 0; p < 16; ++p) ds += ew[n * 16 + p];
    dacc += ds;

    // re-read E in A-fragment layout (m = lane&15, K-dim = p)
    v2f ae[4];
#pragma unroll
    for (int g = 0; g < 4; ++g)
      ae[g] = *(const v2f*)(ew + n * 16 + 4 * g + 2 * hh);

    // ---- out_acc += E @ V : 4 K-groups x 8 d-tiles (32 WMMAs) ----
    // g outer / t inner: 8 consecutive WMMAs share the same A fragment.
#pragma unroll
    for (int g = 0; g < 4; ++g) {
#pragma unroll
      for (int t = 0; t < 8; ++t) {
        v2f b;
        b.x = vbase[(size_t)(4 * g + 2 * hh) * stride + t * 16 + n];
        b.y = vbase[(size_t)(4 * g + 2 * hh + 1) * stride + t * 16 + n];
        af[t] = __builtin_amdgcn_wmma_f32_16x16x4_f32(false, ae[g], false, b,
                                                      (short)0, af[t], false,
                                                      false);
      }
    }
  }

  // merge this wave's register accumulators into the block (ds_add_f32)
#pragma unroll
  for (int t = 0; t < 8; ++t)
#pragma unroll
    for (int r = 0; r < 8; ++r)
      atomicAdd(&acc[(r + 8 * hh) * D_ + t * 16 + n], af[t][r]);
  if (hh == 0) atomicAdd(&den[n], dacc);
  __syncthreads();

  // write block partials to global workspace
  float* pa = pacc + (size_t)blockIdx.x * (M_ * D_);
  for (int i = threadIdx.x; i < M_ * D_; i += 256) pa[i] = acc[i];
  if (threadIdx.x < M_) pden[blockIdx.x * M_ + threadIdx.x] = den[threadIdx.x];
}

// ---------------------------------------------------------------------------
// Kernel 3: combine split partials and emit [M, H*D] fp32 output.
// ---------------------------------------------------------------------------
__global__ __launch_bounds__(256) void attn_combine(
    const float* __restrict__ pacc, const float* __restrict__ pden,
    float* __restrict__ out) {
  const int idx = blockIdx.x * 256 + threadIdx.x;  // 16*4096 = 65536
  const int m   = idx / (H_ * D_);
  const int col = idx % (H_ * D_);
  const int h   = col / D_;
  const int d   = col % D_;
  float a = 0.f, dn = 0.f;
#pragma unroll
  for (int s = 0; s < ASPLIT; ++s) {
    const int b = h * ASPLIT + s;
    a  += pacc[(size_t)b * (M_ * D_) + m * D_ + d];
    dn += pden[b * M_ + m];
  }
  out[idx] = a / dn;
}

// ---------------------------------------------------------------------------
// kernel_launch: X, W_qkv, cache_K, cache_V (all fp32) -> out [16, 4096] fp32
// Workspace (floats): qkv_part | qkv | pacc | pden   (~8.2 MiB total)
// ---------------------------------------------------------------------------
extern "C" void kernel_launch(void* const* d_in, const int* in_sizes, int n_in,
                              void* d_out, int out_size, void* d_ws,
                              size_t ws_size, hipStream_t stream) {
  (void)in_sizes; (void)n_in; (void)out_size; (void)ws_size;
  const float* X  = (const float*)d_in[0];
  const float* W  = (const float*)d_in[1];
  const float* cK = (const float*)d_in[2];
  const float* cV = (const float*)d_in[3];
  float* out = (float*)d_out;

  float* ws       = (float*)d_ws;
  float* qkv_part = ws;                                     // KSPLIT*16*12288
  float* qkv      = qkv_part + (size_t)KSPLIT * M_ * QKVC;  // 16*12288
  float* pacc     = qkv + (size_t)M_ * QKVC;                // 512*2048
  float* pden     = pacc + (size_t)H_ * ASPLIT * M_ * D_;   // 512*16

  // 1) QKV projection (split-K): 768 tiles * 4 splits = 3072 waves
  qkv_gemm_splitk<<<(QKVC / 16) * KSPLIT / 8, 256, 0, stream>>>(X, W, qkv_part);
  // 1b) reduce split-K partials
  qkv_reduce<<<(M_ * QKVC) / 256, 256, 0, stream>>>(qkv_part, qkv);
  // 2) attention partials: 32 heads * 16 splits = 512 blocks
  attn_partial<<<H_ * ASPLIT, 256, 0, stream>>>(qkv, cK, cV, pacc, pden);
  // 3) combine and write output
  attn_combine<<<(M_ * H_ * D_) / 256, 256, 0, stream>>>(pacc, pden, out);
}